// GraphAttentionLearningModule_15771119911348
// MI455X (gfx1250) — compile-verified
//
#include <hip/hip_runtime.h>
#include <hip/hip_bf16.h>
#include <math.h>

typedef __attribute__((ext_vector_type(2))) float v2f;
typedef __attribute__((ext_vector_type(8))) float v8f;

#define Nn 2048
#define Dd 128
#define Hh 4
#define Cc 64
#define HC 256           // Hh * Cc
#define NEG_SLOPE 0.2f

// ---------------------------------------------------------------------------
// Kernel 1: X[2048,256] = input_emb[2048,128] @ W[128,256] via fp32 WMMA.
// One wave per 16x16 output tile; K loop fully unrolled -> 32 chained
// v_wmma_f32_16x16x4_f32 per wave.
//
// f32 16x16x4 fragment layout (ISA 7.12.2):
//   A (16x4):  lane = hi*16 + m;  v0 = A[m][2*hi], v1 = A[m][2*hi+1]
//   B (4x16):  lane = hi*16 + n;  v0 = B[2*hi][n], v1 = B[2*hi+1][n]
//   C/D (16x16): VGPR r, lane = hi*16 + n -> C[r + 8*hi][n]
// ---------------------------------------------------------------------------
__global__ void gat_gemm_wmma(const float* __restrict__ A,
                              const float* __restrict__ B,
                              float* __restrict__ X) {
  int wave = threadIdx.x >> 5;
  int lane = threadIdx.x & 31;
  int tile = blockIdx.x * 8 + wave;   // 2048 tiles total: 128 row x 16 col
  int tm = tile >> 4;                 // row tile  [0,128)
  int tn = tile & 15;                 // col tile  [0,16)
  int m  = lane & 15;                 // also the B/C column index n
  int hi = lane >> 4;

  const float* arow = A + (tm * 16 + m) * Dd;
  const float* bcol = B + tn * 16 + m;

  v8f acc = {};
#pragma unroll
  for (int k = 0; k < Dd; k += 4) {
    int ka = k + 2 * hi;
    v2f a = *(const v2f*)(arow + ka);          // A[m][ka], A[m][ka+1]
    v2f b;
    b.x = bcol[(ka + 0) * HC];                 // B[ka][n]
    b.y = bcol[(ka + 1) * HC];                 // B[ka+1][n]
    acc = __builtin_amdgcn_wmma_f32_16x16x4_f32(
        /*neg_a=*/false, a, /*neg_b=*/false, b,
        /*c_mod=*/(short)0, acc, /*reuse_a=*/false, /*reuse_b=*/false);
  }

  float* xout = X + tn * 16 + m;
#pragma unroll
  for (int r = 0; r < 8; ++r)
    xout[(tm * 16 + r + 8 * hi) * HC] = acc[r];
}

// ---------------------------------------------------------------------------
// Kernel 2: per-node attention coefficients
//   a_src[n,h] = dot(x[n,h,:], att_src[h,:]) ; same for a_dst.
// ---------------------------------------------------------------------------
__global__ void gat_attn_coeff(const float* __restrict__ X,
                               const float* __restrict__ att_src,
                               const float* __restrict__ att_dst,
                               float* __restrict__ a_src,
                               float* __restrict__ a_dst) {
  int idx = blockIdx.x * blockDim.x + threadIdx.x;  // n*Hh + h
  if (idx >= Nn * Hh) return;
  int n = idx >> 2, h = idx & 3;
  const float* xp = X + n * HC + h * Cc;
  const float* as = att_src + h * Cc;
  const float* ad = att_dst + h * Cc;
  float s0 = 0.f, s1 = 0.f;
#pragma unroll 8
  for (int c = 0; c < Cc; ++c) {
    float xv = xp[c];
    s0 += xv * as[c];
    s1 += xv * ad[c];
  }
  a_src[idx] = s0;
  a_dst[idx] = s1;
}

// ---------------------------------------------------------------------------
// Kernel 3: per-(dst,head) softmax stats over the src axis (diagonal masked).
// One block of 128 threads per dst node; online (max,sum) + LDS tree merge.
// ---------------------------------------------------------------------------
__global__ void gat_softmax_stats(const float* __restrict__ a_src,
                                  const float* __restrict__ a_dst,
                                  float* __restrict__ mx,
                                  float* __restrict__ den) {
  __shared__ float sm[Hh][128];
  __shared__ float ss[Hh][128];
  int d = blockIdx.x;
  int tid = threadIdx.x;

  float ad[Hh];
#pragma unroll
  for (int h = 0; h < Hh; ++h) ad[h] = a_dst[d * Hh + h];

  float m[Hh], s[Hh];
#pragma unroll
  for (int h = 0; h < Hh; ++h) { m[h] = -INFINITY; s[h] = 0.f; }

  for (int src = tid; src < Nn; src += 128) {
    if (src == d) continue;  // no self loops
#pragma unroll
    for (int h = 0; h < Hh; ++h) {
      float v = a_src[src * Hh + h] + ad[h];
      v = (v >= 0.f) ? v : NEG_SLOPE * v;      // leaky_relu
      if (v > m[h]) { s[h] = s[h] * __expf(m[h] - v) + 1.f; m[h] = v; }
      else          { s[h] += __expf(v - m[h]); }
    }
  }

#pragma unroll
  for (int h = 0; h < Hh; ++h) { sm[h][tid] = m[h]; ss[h][tid] = s[h]; }
  __syncthreads();

  for (int off = 64; off > 0; off >>= 1) {
    if (tid < off) {
#pragma unroll
      for (int h = 0; h < Hh; ++h) {
        float m1 = sm[h][tid],       s1 = ss[h][tid];
        float m2 = sm[h][tid + off], s2 = ss[h][tid + off];
        float M = fmaxf(m1, m2);
        float S = (M == -INFINITY) ? 0.f
                                   : s1 * __expf(m1 - M) + s2 * __expf(m2 - M);
        sm[h][tid] = M; ss[h][tid] = S;
      }
    }
    __syncthreads();
  }

  if (tid == 0) {
#pragma unroll
    for (int h = 0; h < Hh; ++h) {
      mx[d * Hh + h]  = sm[h][0];
      den[d * Hh + h] = ss[h][0];
    }
  }
}

// ---------------------------------------------------------------------------
// Kernel 4: adj[s,d] = mean_h softmax_s(logits)[s,d,h]; diagonal = 0.
// Row s per blockIdx.y so a_src[s,*] is block-uniform (scalar loads).
// ---------------------------------------------------------------------------
__global__ void gat_adj(const float* __restrict__ a_src,
                        const float* __restrict__ a_dst,
                        const float* __restrict__ mx,
                        const float* __restrict__ den,
                        float* __restrict__ adj) {
  int d = blockIdx.x * blockDim.x + threadIdx.x;
  int s = blockIdx.y;
  if (d >= Nn) return;
  float out = 0.f;
  if (s != d) {
#pragma unroll
    for (int h = 0; h < Hh; ++h) {
      float v = a_src[s * Hh + h] + a_dst[d * Hh + h];
      v = (v >= 0.f) ? v : NEG_SLOPE * v;
      out += __expf(v - mx[d * Hh + h]) / den[d * Hh + h];
    }
    out *= 0.25f;  // mean over Hh = 4 heads
  }
  adj[s * Nn + d] = out;
}

// ---------------------------------------------------------------------------
// Kernel 5: complete-graph edge index (self loops removed), row-major over
// src with diagonal skipped: e -> (s = e/(N-1), d = r + (r>=s)).
// ---------------------------------------------------------------------------
__global__ void gat_edges(float* __restrict__ eout) {
  long long e = (long long)blockIdx.x * blockDim.x + threadIdx.x;
  const long long E = (long long)Nn * (Nn - 1);
  if (e >= E) return;
  int s = (int)(e / (Nn - 1));
  int r = (int)(e - (long long)s * (Nn - 1));
  int dsti = r + (r >= s ? 1 : 0);
  eout[e]     = (float)s;     // edge_index[0, e]
  eout[E + e] = (float)dsti;  // edge_index[1, e]
}

extern "C" void kernel_launch(void* const* d_in, const int* in_sizes, int n_in,
                              void* d_out, int out_size, void* d_ws, size_t ws_size,
                              hipStream_t stream) {
  const float* input_emb = (const float*)d_in[0];  // [2048,128]
  const float* W         = (const float*)d_in[1];  // [128,256]
  const float* att_src   = (const float*)d_in[2];  // [4,64]
  const float* att_dst   = (const float*)d_in[3];  // [4,64]
  float* out = (float*)d_out;

  // workspace layout
  float* X     = (float*)d_ws;          // [2048,256]
  float* a_src = X + Nn * HC;           // [2048,4]
  float* a_dst = a_src + Nn * Hh;       // [2048,4]
  float* mx    = a_dst + Nn * Hh;       // [2048,4]
  float* den   = mx + Nn * Hh;          // [2048,4]

  const long long E = (long long)Nn * (Nn - 1);
  float* eout = out;          // edge_index flat: [2, E]
  float* adj  = out + 2 * E;  // adj: [2048,2048]

  // 2048 tiles / 8 waves per block = 256 blocks
  gat_gemm_wmma<<<256, 256, 0, stream>>>(input_emb, W, X);
  gat_attn_coeff<<<(Nn * Hh + 255) / 256, 256, 0, stream>>>(X, att_src, att_dst,
                                                            a_src, a_dst);
  gat_softmax_stats<<<Nn, 128, 0, stream>>>(a_src, a_dst, mx, den);
  dim3 g4(Nn / 256, Nn);
  gat_adj<<<g4, 256, 0, stream>>>(a_src, a_dst, mx, den, adj);
  gat_edges<<<(int)((E + 255) / 256), 256, 0, stream>>>(eout);
}